// RlSaTree_20358144983213
// MI455X (gfx1250) — compile-verified
//
#include <hip/hip_runtime.h>
#include <hip/hip_bf16.h>

typedef __attribute__((ext_vector_type(2))) float v2f;
typedef __attribute__((ext_vector_type(8))) float v8f;

#define BATCH   16
#define SEQL    32
#define DIM     300
#define HID     300
#define TWOH    600
#define THREEH  900
#define FIVEH   1500
#define NW_PAD  608          // 38 N-tiles of 16, leaf GEMM
#define NC_PAD  1504         // 94 N-tiles of 16, comp GEMM
#define NTILES  (NC_PAD / 16)
#define MAXN    (SEQL - 1)
#define NWG     12           // persistent workgroups (96 wave32 tile slots >= 94)
#define TPB     256

__device__ __forceinline__ float sigm(float x) { return 1.0f / (1.0f + __expf(-x)); }

// ---------------------------------------------------------------------------
// Device-scope grid barrier (agent-scope atomics -> global_wb / global_inv).
__device__ void grid_barrier(int* cnt, int* gen) {
    __threadfence();
    __syncthreads();
    if (threadIdx.x == 0) {
        int g = __hip_atomic_load(gen, __ATOMIC_RELAXED, __HIP_MEMORY_SCOPE_AGENT);
        int prev = __hip_atomic_fetch_add(cnt, 1, __ATOMIC_ACQ_REL, __HIP_MEMORY_SCOPE_AGENT);
        if (prev == NWG - 1) {
            __hip_atomic_store(cnt, 0, __ATOMIC_RELAXED, __HIP_MEMORY_SCOPE_AGENT);
            __hip_atomic_store(gen, g + 1, __ATOMIC_RELEASE, __HIP_MEMORY_SCOPE_AGENT);
        } else {
            while (__hip_atomic_load(gen, __ATOMIC_ACQUIRE, __HIP_MEMORY_SCOPE_AGENT) == g) {
                __builtin_amdgcn_s_sleep(2);
            }
        }
    }
    __syncthreads();
}

// ---------------------------------------------------------------------------
// Weight repack: pair-interleaved WMMA B layout.
// Bp[((k/2)*NPAD + n)*2 + (k&1)] = W[n][k]  (zero-pad n >= N)
__global__ void k_pack_wword(const float* __restrict__ Ww, float* __restrict__ Bwp) {
    int idx = blockIdx.x * blockDim.x + threadIdx.x;
    if (idx >= DIM * NW_PAD) return;
    int p = idx / (NW_PAD * 2), r = idx % (NW_PAD * 2);
    int n = r >> 1, k = p * 2 + (r & 1);
    Bwp[idx] = (n < TWOH) ? Ww[(size_t)n * DIM + k] : 0.0f;
}
__global__ void k_pack_wcomp(const float* __restrict__ Wc, float* __restrict__ Bcp) {
    int idx = blockIdx.x * blockDim.x + threadIdx.x;
    if (idx >= THREEH * NC_PAD) return;
    int p = idx / (NC_PAD * 2), r = idx % (NC_PAD * 2);
    int n = r >> 1, k = p * 2 + (r & 1);
    Bcp[idx] = (n < FIVEH) ? Wc[(size_t)n * THREEH + k] : 0.0f;
}

// ---------------------------------------------------------------------------
// Leaf GEMM: state(512 x 600) = X @ W_word^T + b_word. One wave per 16x16 tile,
// 75 chained V_WMMA_F32_16X16X4_F32; A via global b64, B via packed-pair b64.
__global__ void k_leaf_gemm(const float* __restrict__ X, const float* __restrict__ Bwp,
                            const float* __restrict__ bw, float* __restrict__ state) {
    int mt   = blockIdx.x & 31;
    int nt   = blockIdx.x >> 5;
    int lane = threadIdx.x;
    int half = lane >> 4;
    int lid  = lane & 15;
    int m0 = mt * 16, n0 = nt * 16;
    const v2f* Arow2 = (const v2f*)(X + (size_t)(m0 + lid) * DIM);
    const v2f* Bp2   = (const v2f*)Bwp;
    v8f acc = {};
    for (int kk = 0; kk < DIM; kk += 4) {
        int p = (kk >> 1) + half;
        v2f a = Arow2[p];
        v2f b = Bp2[(size_t)p * NW_PAD + n0 + lid];
        acc = __builtin_amdgcn_wmma_f32_16x16x4_f32(false, a, false, b,
                                                    (short)0, acc, false, false);
    }
    int n = n0 + lid;
    if (n < TWOH) {
        float bias = bw[n];
        for (int g = 0; g < 8; ++g) {
            int m = m0 + g + 8 * half;
            state[(size_t)m * TWOH + n] = acc[g] + bias;
        }
    }
}

// ---------------------------------------------------------------------------
// Rank MLP scores (block per token).
__global__ void k_scores(const float* __restrict__ X, const float* __restrict__ W1,
                         const float* __restrict__ W2, float* __restrict__ scores) {
    __shared__ float red[128];
    int bl = blockIdx.x;
    int j  = threadIdx.x;
    const float* x = X  + (size_t)bl * DIM;
    const float* w = W1 + (size_t)j * DIM;
    float acc = 0.0f;
    for (int k = 0; k < DIM; ++k) acc += x[k] * w[k];
    red[j] = fmaxf(acc, 0.0f) * W2[j];
    __syncthreads();
    for (int off = 64; off > 0; off >>= 1) {
        if (j < off) red[j] += red[j + off];
        __syncthreads();
    }
    if (j == 0) scores[bl] = red[0];
}

// ---------------------------------------------------------------------------
// Greedy post-order tree build + log-prob terms; also resets grid-barrier state.
// Slot: -1 empty, 0..31 leaf, 32+k internal node k.
__global__ void k_build(const float* __restrict__ scores, const int* __restrict__ length,
                        int* __restrict__ meta, int* __restrict__ cnt,
                        float* __restrict__ logp, int* __restrict__ bar) {
    int i = blockIdx.x * blockDim.x + threadIdx.x;
    if (i == 0) { bar[0] = 0; bar[1] = 0; }
    if (i >= BATCH) return;
    const float* sc = scores + i * SEQL;
    int len = length[i];
    if (len > SEQL) len = SEQL;
    if (len < 0) len = 0;

    int st_s[40], st_e[40], st_p[40], st_pos[40], st_l[40];
    int sp = 0, count = 0, ret = -1;
    st_s[0] = 0; st_e[0] = len; st_p[0] = 0; sp = 1;
    while (sp > 0) {
        int t = sp - 1;
        int ph = st_p[t];
        if (ph == 0) {
            int s0 = st_s[t], e0 = st_e[t];
            if (e0 <= s0)     { ret = -1; sp--; continue; }
            if (e0 == s0 + 1) { ret = s0; sp--; continue; }
            int pos = s0; float best = sc[s0];
            for (int j = s0 + 1; j < e0; ++j) {
                float v = sc[j];
                if (v > best) { best = v; pos = j; }
            }
            st_pos[t] = pos; st_p[t] = 1;
            st_s[sp] = s0; st_e[sp] = pos; st_p[sp] = 0; sp++;
        } else if (ph == 1) {
            st_l[t] = ret; st_p[t] = 2;
            st_s[sp] = st_pos[t] + 1; st_e[sp] = st_e[t]; st_p[sp] = 0; sp++;
        } else {
            int s0 = st_s[t], e0 = st_e[t], pos = st_pos[t];
            int k = count++;
            float mx = sc[s0];
            for (int j = s0 + 1; j < e0; ++j) mx = fmaxf(mx, sc[j]);
            float sum = 0.0f;
            for (int j = s0; j < e0; ++j) sum += __expf(sc[j] - mx);
            float lz = mx + __logf(sum);
            logp[i * MAXN + k] = (float)(e0 - s0) * (sc[pos] - lz);
            meta[(i * MAXN + k) * 3 + 0] = pos;
            meta[(i * MAXN + k) * 3 + 1] = st_l[t];
            meta[(i * MAXN + k) * 3 + 2] = ret;
            ret = SEQL + k;
            sp--;
        }
    }
    cnt[i] = count;
}

// ---------------------------------------------------------------------------
// Persistent fused tree evaluation: all 31 post-order steps in ONE launch.
// Per step: assemble A(16x900) into LDS -> per-wave WMMA GEMM tile (A from
// LDS via ds_load_b64, B via packed-pair global b64) -> grid barrier ->
// LSTM gates -> grid barrier. Final gather done by WG 0.
__global__ void k_tree_eval(const int* __restrict__ meta, const int* __restrict__ cnt,
                            const float* __restrict__ state, const float* __restrict__ Bcp,
                            const float* __restrict__ bc, const float* __restrict__ logp,
                            float* __restrict__ V, float* __restrict__ nodeH,
                            float* __restrict__ nodeC, float* __restrict__ out,
                            int* __restrict__ bar) {
    __shared__ float As[BATCH * THREEH];          // 57.6 KB of the 320 KB WGP LDS
    int tid   = threadIdx.x;
    int waveId = tid >> 5;
    int lane   = tid & 31;
    int half   = lane >> 4;
    int lid    = lane & 15;
    int* bcnt = bar;
    int* bgen = bar + 1;

    for (int s = 0; s < MAXN; ++s) {
        // ---- Phase A: gather [hl | hx | hr] rows into LDS (zeros if inactive)
        for (int idx = tid; idx < BATCH * THREEH; idx += TPB) {
            int r = idx / THREEH, c = idx % THREEH;
            float v = 0.0f;
            if (s < cnt[r]) {
                int slot;
                int col;
                if (c < HID)            { slot = meta[(r * MAXN + s) * 3 + 1]; col = c; }
                else if (c < 2 * HID)   { slot = meta[(r * MAXN + s) * 3 + 0]; col = c - HID; }
                else                    { slot = meta[(r * MAXN + s) * 3 + 2]; col = c - 2 * HID; }
                if (c >= HID && c < 2 * HID) {   // hx: slot is a leaf position
                    v = state[((size_t)r * SEQL + slot) * TWOH + col];
                } else if (slot >= 0) {
                    v = (slot < SEQL)
                        ? state[((size_t)r * SEQL + slot) * TWOH + col]
                        : nodeH[((size_t)r * MAXN + (slot - SEQL)) * HID + col];
                }
            }
            As[idx] = v;
        }
        __syncthreads();

        // ---- Phase B: WMMA GEMM, one 16x16 tile per wave (wave-uniform guard)
        int tile = blockIdx.x * (TPB / 32) + waveId;
        if (tile < NTILES) {
            int n0 = tile * 16;
            const v2f* Arow2 = (const v2f*)(As + (size_t)lid * THREEH);
            const v2f* Bp2   = (const v2f*)Bcp;
            v8f acc = {};
            for (int kk = 0; kk < THREEH; kk += 4) {
                int p = (kk >> 1) + half;
                v2f a = Arow2[p];
                v2f b = Bp2[(size_t)p * NC_PAD + n0 + lid];
                acc = __builtin_amdgcn_wmma_f32_16x16x4_f32(false, a, false, b,
                                                            (short)0, acc, false, false);
            }
            int n = n0 + lid;
            float bias = (n < FIVEH) ? bc[n] : 0.0f;
            for (int g = 0; g < 8; ++g) {
                int m = g + 8 * half;                   // sentence
                V[(size_t)m * NC_PAD + n] = acc[g] + bias;
            }
        }
        grid_barrier(bcnt, bgen);

        // ---- Phase C: LSTM gates, distributed over 16x300 elements
        for (int idx = blockIdx.x * TPB + tid; idx < BATCH * HID; idx += NWG * TPB) {
            int i = idx / HID, n = idx % HID;
            if (s < cnt[i]) {
                int ls = meta[(i * MAXN + s) * 3 + 1];
                int rs = meta[(i * MAXN + s) * 3 + 2];
                const float* Vi = V + (size_t)i * NC_PAD;
                float gi = Vi[n];
                float fl = Vi[HID + n];
                float fr = Vi[2 * HID + n];
                float u  = Vi[3 * HID + n];
                float o  = Vi[4 * HID + n];
                float cl = 0.0f, cr = 0.0f;
                if (ls >= 0)
                    cl = (ls < SEQL) ? state[((size_t)i * SEQL + ls) * TWOH + HID + n]
                                     : nodeC[((size_t)i * MAXN + (ls - SEQL)) * HID + n];
                if (rs >= 0)
                    cr = (rs < SEQL) ? state[((size_t)i * SEQL + rs) * TWOH + HID + n]
                                     : nodeC[((size_t)i * MAXN + (rs - SEQL)) * HID + n];
                float c = cl * sigm(fl + 1.0f) + cr * sigm(fr + 1.0f) + tanhf(u) * sigm(gi);
                float h = sigm(o) * tanhf(c);
                nodeH[((size_t)i * MAXN + s) * HID + n] = h;
                nodeC[((size_t)i * MAXN + s) * HID + n] = c;
            }
        }
        grid_barrier(bcnt, bgen);
    }

    // ---- Final gather: out[i] = [root_h, root_c, sum(logp)]
    if (blockIdx.x == 0) {
        for (int idx = tid; idx < BATCH * HID; idx += TPB) {
            int i = idx / HID, n = idx % HID;
            int c = cnt[i];
            float h, cc;
            if (c > 0) {
                h  = nodeH[((size_t)i * MAXN + c - 1) * HID + n];
                cc = nodeC[((size_t)i * MAXN + c - 1) * HID + n];
            } else {
                h  = state[(size_t)i * SEQL * TWOH + n];
                cc = state[(size_t)i * SEQL * TWOH + HID + n];
            }
            out[(size_t)i * (TWOH + 1) + n]       = h;
            out[(size_t)i * (TWOH + 1) + HID + n] = cc;
        }
        for (int i = tid; i < BATCH; i += TPB) {
            float sum = 0.0f;
            int c = cnt[i];
            for (int k = 0; k < c; ++k) sum += logp[i * MAXN + k];
            out[(size_t)i * (TWOH + 1) + TWOH] = sum;
        }
    }
}

// ---------------------------------------------------------------------------
extern "C" void kernel_launch(void* const* d_in, const int* in_sizes, int n_in,
                              void* d_out, int out_size, void* d_ws, size_t ws_size,
                              hipStream_t stream) {
    const float* X      = (const float*)d_in[0];   // (16,32,300)
    // d_in[1] sentence_word: unused by the reference computation
    const int*   length = (const int*)  d_in[2];   // (16,)
    const float* Ww     = (const float*)d_in[3];   // (600,300)
    const float* bw     = (const float*)d_in[4];   // (600,)
    const float* W1     = (const float*)d_in[5];   // (128,300)
    const float* W2     = (const float*)d_in[6];   // (1,128)
    const float* Wc     = (const float*)d_in[7];   // (1500,900)
    const float* bc     = (const float*)d_in[8];   // (1500,)
    float* out = (float*)d_out;                    // (16, 601)

    float* ws = (float*)d_ws;
    size_t off = 0;
    float* state  = ws + off; off += (size_t)BATCH * SEQL * TWOH;   // 307200
    float* Bwp    = ws + off; off += (size_t)DIM * NW_PAD;          // 182400
    float* Bcp    = ws + off; off += (size_t)THREEH * NC_PAD;       // 1353600
    float* scores = ws + off; off += (size_t)BATCH * SEQL;          // 512
    float* Vbuf   = ws + off; off += (size_t)BATCH * NC_PAD;        // 24064
    float* nodeH  = ws + off; off += (size_t)BATCH * MAXN * HID;    // 148800
    float* nodeC  = ws + off; off += (size_t)BATCH * MAXN * HID;    // 148800
    float* logp   = ws + off; off += (size_t)BATCH * MAXN;          // 496
    int*   meta   = (int*)(ws + off); off += (size_t)BATCH * MAXN * 3;
    int*   cnt    = (int*)(ws + off); off += BATCH;
    int*   bar    = (int*)(ws + off); off += 2;

    // Pair-interleaved weight packing (one-time, L2-resident afterwards)
    k_pack_wword<<<(DIM * NW_PAD + 255) / 256, 256, 0, stream>>>(Ww, Bwp);
    k_pack_wcomp<<<(THREEH * NC_PAD + 255) / 256, 256, 0, stream>>>(Wc, Bcp);

    // Leaf states via f32 WMMA
    k_leaf_gemm<<<32 * 38, 32, 0, stream>>>(X, Bwp, bw, state);

    // Rank scores
    k_scores<<<BATCH * SEQL, 128, 0, stream>>>(X, W1, W2, scores);

    // Tree structure + log-prob terms + barrier reset
    k_build<<<1, 32, 0, stream>>>(scores, length, meta, cnt, logp, bar);

    // Single persistent launch for all 31 lockstep post-order steps + gather
    k_tree_eval<<<NWG, TPB, 0, stream>>>(meta, cnt, state, Bcp, bc, logp,
                                         Vbuf, nodeH, nodeC, out, bar);
}